// VectorQuantizer_83743272337531
// MI455X (gfx1250) — compile-verified
//
#include <hip/hip_runtime.h>
#include <math.h>

typedef float v2f __attribute__((ext_vector_type(2)));
typedef float v8f __attribute__((ext_vector_type(8)));

// Problem constants (B=32, D=64, H=32, W=32, K=1024)
#define NPIX   32768       // B*H*W
#define DDIM   64
#define KCODE  1024
#define HW     1024        // H*W
#define COMMIT 0.25f

// d_out layout (floats): zq[2097152] | loss[1] | perp[1] | idx[32768] | dist[33554432]
#define OFF_LOSS 2097152
#define OFF_PERP 2097153
#define OFF_IDX  2097154
#define OFF_DIST 2129922

// ---------------------------------------------------------------------------
// Kernel 1: f2[n] = sum_d z_e[n,d]^2 ; e2[k] = sum_d emb[k,d]^2 ; zero counts
// ---------------------------------------------------------------------------
__global__ void __launch_bounds__(256)
vq_prep_kernel(const float* __restrict__ ze, const float* __restrict__ emb,
               float* __restrict__ f2, float* __restrict__ e2,
               unsigned* __restrict__ counts)
{
    int t = blockIdx.x * 256 + threadIdx.x;
    if (t < NPIX) {
        int b = t >> 10, hw = t & (HW - 1);
        const float* zb = ze + (size_t)b * DDIM * HW + hw;
        float s = 0.f;
#pragma unroll 8
        for (int d = 0; d < DDIM; ++d) { float z = zb[(size_t)d * HW]; s += z * z; }
        f2[t] = s;
    } else if (t < NPIX + KCODE) {
        int k = t - NPIX;
        const float4* er = (const float4*)(emb + (size_t)k * DDIM);
        float s = 0.f;
#pragma unroll
        for (int i = 0; i < DDIM / 4; ++i) {
            float4 v = er[i];
            s += v.x * v.x + v.y * v.y + v.z * v.z + v.w * v.w;
        }
        e2[k] = s;
    }
    if (t < KCODE) counts[t] = 0u;
}

// ---------------------------------------------------------------------------
// Kernel 2: distances [N,K] via V_WMMA_F32_16X16X4_F32 + fused row argmin.
// One wave per 16-row tile of N; A panel (16x64 fp32) kept in registers,
// B fragments loaded from L2-resident embedding. dist = f2 + e2 - 2*dot.
// ---------------------------------------------------------------------------
__global__ void __launch_bounds__(256)
vq_dist_kernel(const float* __restrict__ ze, const float* __restrict__ emb,
               const float* __restrict__ f2, const float* __restrict__ e2,
               float* __restrict__ dist, int* __restrict__ idx_i,
               float* __restrict__ idx_f)
{
    const int wave = threadIdx.x >> 5;
    const int lane = threadIdx.x & 31;
    const int col  = lane & 15;      // N-column within tile / A row M
    const int hi   = lane >> 4;      // half-wave select
    const int n0   = (blockIdx.x * 8 + wave) * 16;   // tile base row (0..32752)
    const int b    = n0 >> 10;
    const int hw   = (n0 & (HW - 1)) + col;          // this lane's A row position

    // --- A fragments: a[s] covers GEMM-K (= embed dim d) slice [4s, 4s+3] ---
    // layout: lanes 0-15 hold K={4s,4s+1}, lanes 16-31 hold K={4s+2,4s+3}
    const float* zb = ze + (size_t)b * DDIM * HW;
    v2f a[16];
#pragma unroll
    for (int s = 0; s < 16; ++s) {
        int d0 = 4 * s + 2 * hi;
        a[s].x = zb[(size_t)d0 * HW + hw];
        a[s].y = zb[(size_t)(d0 + 1) * HW + hw];
    }
    // C/D layout: VGPR v, lanes 0-15 -> row v, lanes 16-31 -> row v+8
    float f2r[8];
#pragma unroll
    for (int v = 0; v < 8; ++v) f2r[v] = f2[n0 + v + 8 * hi];

    float best[8]; int bi[8];
#pragma unroll
    for (int v = 0; v < 8; ++v) { best[v] = 3.4e38f; bi[v] = 0; }

    for (int kt = 0; kt < KCODE / 16; ++kt) {
        const int k0 = kt * 16;
        // B fragment source: emb row (k0+col), this half-wave reads d = 4s+2hi..+1
        const float* brow = emb + (size_t)(k0 + col) * DDIM + 2 * hi;
        v8f c = {};
#pragma unroll
        for (int s = 0; s < 16; ++s) {
            v2f bf = *(const v2f*)(brow + 4 * s);
            c = __builtin_amdgcn_wmma_f32_16x16x4_f32(
                    false, a[s], false, bf, (short)0, c, false, false);
        }
        const float e2c = e2[k0 + col];
        float* drow = dist + (size_t)n0 * KCODE + k0 + col;
#pragma unroll
        for (int v = 0; v < 8; ++v) {
            int   row  = v + 8 * hi;
            float dval = f2r[v] + e2c - 2.0f * c[v];
            drow[(size_t)row * KCODE] = dval;
            if (dval < best[v]) { best[v] = dval; bi[v] = k0 + col; }
        }
    }
    // argmin reduce across the 16 lanes sharing each row (first-index ties)
#pragma unroll
    for (int off = 1; off < 16; off <<= 1) {
#pragma unroll
        for (int v = 0; v < 8; ++v) {
            float ov = __shfl_xor(best[v], off, 32);
            int   oi = __shfl_xor(bi[v],   off, 32);
            if (ov < best[v] || (ov == best[v] && oi < bi[v])) {
                best[v] = ov; bi[v] = oi;
            }
        }
    }
    if (col == 0) {
#pragma unroll
        for (int v = 0; v < 8; ++v) {
            int row = v + 8 * hi;
            idx_i[n0 + row] = bi[v];
            idx_f[n0 + row] = (float)bi[v];
        }
    }
}

// ---------------------------------------------------------------------------
// Kernel 3: z_q (NCHW), SSE partials (deterministic), index histogram
// ---------------------------------------------------------------------------
__global__ void __launch_bounds__(256)
vq_gather_kernel(const float* __restrict__ ze, const float* __restrict__ emb,
                 const int* __restrict__ idx, float* __restrict__ zq,
                 unsigned* __restrict__ counts, float* __restrict__ partials)
{
    const int t  = blockIdx.x * 256 + threadIdx.x;   // pixel id 0..32767
    const int b  = t >> 10, hw = t & (HW - 1);
    const int k  = idx[t];
    const float* er = emb + (size_t)k * DDIM;
    const float* zb = ze + (size_t)b * DDIM * HW + hw;
    float*       ob = zq + (size_t)b * DDIM * HW + hw;
    float acc = 0.f;
#pragma unroll 8
    for (int d = 0; d < DDIM; ++d) {
        float q = er[d];
        float z = zb[(size_t)d * HW];
        ob[(size_t)d * HW] = q;
        float df = q - z;
        acc += df * df;
    }
    atomicAdd(&counts[k], 1u);
    // block-deterministic SSE reduction
#pragma unroll
    for (int off = 16; off > 0; off >>= 1) acc += __shfl_xor(acc, off, 32);
    __shared__ float wsum[8];
    if ((threadIdx.x & 31) == 0) wsum[threadIdx.x >> 5] = acc;
    __syncthreads();
    if (threadIdx.x == 0) {
        float s = 0.f;
#pragma unroll
        for (int w = 0; w < 8; ++w) s += wsum[w];
        partials[blockIdx.x] = s;
    }
}

// ---------------------------------------------------------------------------
// Kernel 4: perplexity + loss scalars
// ---------------------------------------------------------------------------
__global__ void __launch_bounds__(1024)
vq_final_kernel(const unsigned* __restrict__ counts,
                const float* __restrict__ partials,
                float* __restrict__ out_loss, float* __restrict__ out_perp)
{
    __shared__ float red[32];
    const int t = threadIdx.x;              // 0..1023 == codebook entry
    float p    = (float)counts[t] * (1.0f / (float)NPIX);
    float term = p * logf(p + 1e-10f);
#pragma unroll
    for (int off = 16; off > 0; off >>= 1) term += __shfl_xor(term, off, 32);
    if ((t & 31) == 0) red[t >> 5] = term;
    __syncthreads();
    if (t < 32) {
        float v = red[t];
#pragma unroll
        for (int off = 16; off > 0; off >>= 1) v += __shfl_xor(v, off, 32);
        if (t == 0) {
            float sse = 0.f;
            for (int i = 0; i < 128; ++i) sse += partials[i];
            // codebook_loss + COMMIT*commitment_loss; values are identical
            *out_loss = (1.0f + COMMIT) * sse * (1.0f / ((float)NPIX * (float)DDIM));
            *out_perp = expf(-v);
        }
    }
}

// ---------------------------------------------------------------------------
extern "C" void kernel_launch(void* const* d_in, const int* in_sizes, int n_in,
                              void* d_out, int out_size, void* d_ws, size_t ws_size,
                              hipStream_t stream)
{
    const float* ze  = (const float*)d_in[0];   // (32,64,32,32)
    const float* emb = (const float*)d_in[1];   // (1024,64)
    float* out = (float*)d_out;

    float* zq   = out;
    float* loss = out + OFF_LOSS;
    float* perp = out + OFF_PERP;
    float* idxf = out + OFF_IDX;
    float* dist = out + OFF_DIST;

    float*    wsf      = (float*)d_ws;
    float*    f2       = wsf;                       // 32768
    float*    e2       = wsf + 32768;               // 1024
    int*      idxi     = (int*)(wsf + 33792);       // 32768
    unsigned* counts   = (unsigned*)(wsf + 66560);  // 1024
    float*    partials = wsf + 67584;               // 128

    vq_prep_kernel  <<<(NPIX + KCODE + 255) / 256, 256, 0, stream>>>(ze, emb, f2, e2, counts);
    vq_dist_kernel  <<<(NPIX / 16) / 8, 256, 0, stream>>>(ze, emb, f2, e2, dist, idxi, idxf);
    vq_gather_kernel<<<NPIX / 256, 256, 0, stream>>>(ze, emb, idxi, zq, counts, partials);
    vq_final_kernel <<<1, 1024, 0, stream>>>(counts, partials, loss, perp);
}